// Involution1d_26405458935972
// MI455X (gfx1250) — compile-verified
//
#include <hip/hip_runtime.h>

// ---------------------------------------------------------------------------
// Involution1d fused kernel for MI455X (gfx1250, wave32, WMMA).
//
// out[b,c,l] = sum_k kern[b, g(c)*7+k, l] * x[b, c, l+k-3]
//   h    = W1[64,256]  @ x[256,L]  + b1     (per batch, GEMM over L columns)
//   kern = W2[112,64]  @ h[64,L]   + b2
//
// Fully fused: x read once, out written once (~67 MB total @ 23.3 TB/s).
// GEMMs use V_WMMA_F32_16X16X4_F32; LDS staging uses the gfx1250 async
// global->LDS copy path (ASYNCcnt) when the toolchain exposes it.
// ---------------------------------------------------------------------------

typedef __attribute__((ext_vector_type(2))) float v2f;
typedef __attribute__((ext_vector_type(8))) float v8f;

#if __has_builtin(__builtin_amdgcn_global_load_async_to_lds_b32) && \
    __has_builtin(__builtin_amdgcn_global_load_async_to_lds_b128) && \
    __has_builtin(__builtin_amdgcn_s_wait_asynccnt)
#define USE_ASYNC_LDS 1
typedef int v4i __attribute__((vector_size(16)));
typedef __attribute__((address_space(1))) int* gp32_t;
typedef __attribute__((address_space(3))) int* lp32_t;
typedef __attribute__((address_space(1))) v4i* gp128_t;
typedef __attribute__((address_space(3))) v4i* lp128_t;
#else
#define USE_ASYNC_LDS 0
#endif

constexpr int BATCH = 4;
constexpr int CCH   = 256;   // input channels
constexpr int LLEN  = 8192;  // sequence length
constexpr int CR    = 64;    // C / R
constexpr int GRP   = 16;    // groups
constexpr int KW    = 7;     // kernel width
constexpr int GK    = GRP * KW;  // 112
constexpr int PADW  = 3;     // (K-1)/2

constexpr int LTILE = 64;    // l-positions per workgroup

// LDS strides (floats). Multiples of 4 keep B128 async stores 16B-aligned;
// stride = 4 mod 64 keeps A-fragment row gathers bank-conflict-free.
constexpr int XS  = 72;      // x tile row stride (cols 0..69 = LTILE+2*PADW)
constexpr int WS1 = 260;     // W1 row stride
constexpr int HS  = 68;      // h row stride
constexpr int WS2 = 68;      // W2 row stride
constexpr int KS  = 68;      // kern row stride

// LDS region layout (floats). W1 region is later reused for W2 + kern.
constexpr int X_OFF   = 0;                       // 256*72      = 18432
constexpr int H_OFF   = CCH * XS;                // 18432
constexpr int C_OFF   = H_OFF + CR * HS;         // 22784  (region C)
constexpr int W1_OFF  = C_OFF;                   // 64*260      = 16640 floats
constexpr int W2_OFF  = C_OFF;                   // 112*68      =  7616 floats
constexpr int KRN_OFF = C_OFF + GK * WS2;        // 30400, kern = 112*68 = 7616
constexpr int SMEM_FLOATS = C_OFF + CR * WS1;    // 39424 floats = 154 KB

__device__ __forceinline__ void wait_async_then_barrier() {
#if USE_ASYNC_LDS
    __builtin_amdgcn_s_wait_asynccnt(0);
#endif
    __syncthreads();
}

__global__ __launch_bounds__(256)
void involution1d_kernel(const float* __restrict__ x,
                         const float* __restrict__ W1,
                         const float* __restrict__ b1,
                         const float* __restrict__ W2,
                         const float* __restrict__ b2,
                         float* __restrict__ out)
{
    __shared__ __align__(16) float smem[SMEM_FLOATS];

    const int tid  = threadIdx.x;
    const int l0   = blockIdx.x * LTILE;
    const int b    = blockIdx.y;
    const int wave = tid >> 5;
    const int lane = tid & 31;
    const int lo   = lane & 15;        // row/col within 16
    const int hi   = (lane >> 4) & 1;  // half-wave select (K-pair / M+8 split)

    float* xs  = smem + X_OFF;
    float* hs  = smem + H_OFF;
    float* w1s = smem + W1_OFF;
    float* w2s = smem + W2_OFF;
    float* kns = smem + KRN_OFF;

    // ---------- Phase 1: stage x tile (halo +-3, zero padded) and W1 --------
    {
        const float* xb = x + b * (CCH * LLEN);
        constexpr int XW = LTILE + 2 * PADW;  // 70
        for (int i = tid; i < CCH * XW; i += 256) {
            int r = i / XW;
            int j = i - r * XW;
            int gl = l0 - PADW + j;
            if (gl >= 0 && gl < LLEN) {
#if USE_ASYNC_LDS
                __builtin_amdgcn_global_load_async_to_lds_b32(
                    (gp32_t)(xb + r * LLEN + gl),
                    (lp32_t)(xs + r * XS + j), 0, 0);
#else
                xs[r * XS + j] = xb[r * LLEN + gl];
#endif
            } else {
                xs[r * XS + j] = 0.0f;
            }
        }
        // W1: 64x256 contiguous, 4-float chunks (16B aligned both sides)
        for (int i = tid; i < (CR * CCH) / 4; i += 256) {
            int r = i >> 6;
            int c = (i & 63) << 2;
#if USE_ASYNC_LDS
            __builtin_amdgcn_global_load_async_to_lds_b128(
                (gp128_t)(W1 + i * 4),
                (lp128_t)(w1s + r * WS1 + c), 0, 0);
#else
            {
                const float4 v = *(const float4*)(W1 + i * 4);
                *(float4*)(w1s + r * WS1 + c) = v;
            }
#endif
        }
    }
    wait_async_then_barrier();

    // ---------- Phase 2: h[64 x 64] = W1 @ x_tile + b1 (fp32 WMMA) ----------
    // A 16x4 layout: lanes 0-15 row M=lane {K=k,k+1}; lanes 16-31 {K=k+2,k+3}
    // B 4x16 layout: mirrored; C/D 16x16: vgpr v -> row v (+8 for hi half)
    for (int t = wave; t < 16; t += 8) {
        const int m0 = (t >> 2) << 4;
        const int n0 = (t & 3) << 4;
        v8f acc = {0.f, 0.f, 0.f, 0.f, 0.f, 0.f, 0.f, 0.f};
        const float* arow = w1s + (m0 + lo) * WS1 + (hi ? 2 : 0);
        const float* bcol = xs + PADW + n0 + lo + (hi ? 2 * XS : 0);
#pragma unroll 4
        for (int kk = 0; kk < CCH; kk += 4) {
            v2f a, bf;
            a.x  = arow[kk];
            a.y  = arow[kk + 1];
            bf.x = bcol[kk * XS];
            bf.y = bcol[kk * XS + XS];
            acc = __builtin_amdgcn_wmma_f32_16x16x4_f32(
                false, a, false, bf, (short)0, acc, false, false);
        }
#pragma unroll
        for (int v = 0; v < 8; ++v) {
            const int row = m0 + v + (hi ? 8 : 0);
            hs[row * HS + n0 + lo] = acc[v] + b1[row];
        }
    }
    __syncthreads();

    // ---------- Phase 3a: reload region C with W2 (W1 no longer needed) -----
    for (int i = tid; i < (GK * CR) / 4; i += 256) {
        int r = i >> 4;
        int c = (i & 15) << 2;
#if USE_ASYNC_LDS
        __builtin_amdgcn_global_load_async_to_lds_b128(
            (gp128_t)(W2 + i * 4),
            (lp128_t)(w2s + r * WS2 + c), 0, 0);
#else
        {
            const float4 v = *(const float4*)(W2 + i * 4);
            *(float4*)(w2s + r * WS2 + c) = v;
        }
#endif
    }
    wait_async_then_barrier();

    // ---------- Phase 3b: kern[112 x 64] = W2 @ h + b2 (fp32 WMMA) ----------
    for (int t = wave; t < 28; t += 8) {
        const int m0 = (t >> 2) << 4;   // 0..96
        const int n0 = (t & 3) << 4;
        v8f acc = {0.f, 0.f, 0.f, 0.f, 0.f, 0.f, 0.f, 0.f};
        const float* arow = w2s + (m0 + lo) * WS2 + (hi ? 2 : 0);
        const float* bcol = hs + n0 + lo + (hi ? 2 * HS : 0);
#pragma unroll 4
        for (int kk = 0; kk < CR; kk += 4) {
            v2f a, bf;
            a.x  = arow[kk];
            a.y  = arow[kk + 1];
            bf.x = bcol[kk * HS];
            bf.y = bcol[kk * HS + HS];
            acc = __builtin_amdgcn_wmma_f32_16x16x4_f32(
                false, a, false, bf, (short)0, acc, false, false);
        }
#pragma unroll
        for (int v = 0; v < 8; ++v) {
            const int row = m0 + v + (hi ? 8 : 0);
            kns[row * KS + n0 + lo] = acc[v] + b2[row];
        }
    }
    __syncthreads();

    // ---------- Phase 4: involution (7-tap) + coalesced store ---------------
    {
        const int jc = tid & 63;        // l-column within tile
        const int c0 = tid >> 6;        // starting channel (0..3)
        float* op = out + b * (CCH * LLEN) + l0 + jc;
        for (int c = c0; c < CCH; c += 4) {
            const int g = c >> 4;
            const float* kr = kns + (g * KW) * KS + jc;
            const float* xr = xs + c * XS + jc;
            float acc = 0.0f;
#pragma unroll
            for (int k = 0; k < KW; ++k)
                acc = fmaf(kr[k * KS], xr[k], acc);
            op[c * LLEN] = acc;
        }
    }
}

extern "C" void kernel_launch(void* const* d_in, const int* in_sizes, int n_in,
                              void* d_out, int out_size, void* d_ws, size_t ws_size,
                              hipStream_t stream) {
    const float* x  = (const float*)d_in[0];
    const float* W1 = (const float*)d_in[1];
    const float* b1 = (const float*)d_in[2];
    const float* W2 = (const float*)d_in[3];
    const float* b2 = (const float*)d_in[4];
    float* out = (float*)d_out;

    dim3 grid(LLEN / LTILE, BATCH);   // 128 x 4 = 512 workgroups
    dim3 block(256);                  // 8 wave32
    involution1d_kernel<<<grid, block, 0, stream>>>(x, W1, b1, W2, b2, out);
}